// GraphLearn_88476326298262
// MI455X (gfx1250) — compile-verified
//
#include <hip/hip_runtime.h>
#include <math.h>

typedef __attribute__((ext_vector_type(2))) float v2f;
typedef __attribute__((ext_vector_type(8))) float v8f;

#define B_    4
#define T_    12
#define V_    207
#define F_    32
#define SLABS (B_ * T_)          // 48
#define VT    13                 // ceil(207/16)
#define NROWS (SLABS * V_)       // 9936
#define ALPHA 0.1f
#define EPS_  1e-12f

// ---------------- init: zero per-row accumulators + the two scalar losses ---
__global__ void gl_init_kernel(float* __restrict__ rowsum,
                               float* __restrict__ numer,
                               float* __restrict__ sqsum,
                               float* __restrict__ acc) {
    int i = blockIdx.x * blockDim.x + threadIdx.x;
    if (i < NROWS) { rowsum[i] = 0.f; numer[i] = 0.f; sqsum[i] = 0.f; }
    if (i == 0)    { acc[0] = 0.f; acc[1] = 0.f; }
}

// ---------------- pass 1: one wave per 16x16 tile --------------------------
// WMMA builds the Gram tile dot[i,j] = x_i . x_j  (f32, K=32 via 8 chained
// 16x16x4 f32 WMMAs). VALU loop builds w[i,j] = sum_f |x_i-x_j|*a and the
// row norms; diff_sq = ni + nj - 2*dot.
__global__ __launch_bounds__(32)
void gl_tile_kernel(const float* __restrict__ x,   // [SLABS, V_, F_]
                    const float* __restrict__ a,   // [F_]
                    float* __restrict__ Snum,      // [SLABS, V_, V_] (d_out)
                    float* __restrict__ rowsum,
                    float* __restrict__ numer,
                    float* __restrict__ sqsum) {
    const int lane = threadIdx.x;        // 0..31
    const int half = lane >> 4;          // 0 or 1
    const int n    = lane & 15;

    const int bid  = blockIdx.x;
    const int slab = bid / (VT * VT);
    const int tt   = bid % (VT * VT);
    const int tm   = tt / VT;
    const int tn   = tt % VT;
    const int rowbase = tm * 16;
    const int colbase = tn * 16;

    const float* __restrict__ xs = x + (size_t)slab * V_ * F_;

    // weights a[] in registers
    float av[F_];
#pragma unroll
    for (int f = 0; f < F_; ++f) av[f] = a[f];

    // this lane's column vector (fixed across the tile's 8 rows)
    const int j  = colbase + n;
    const int jc = (j < V_) ? j : (V_ - 1);
    float xjv[F_];
#pragma unroll
    for (int f = 0; f < F_; ++f) xjv[f] = xs[(size_t)jc * F_ + f];
    float nj = 0.f;
#pragma unroll
    for (int f = 0; f < F_; ++f) nj = fmaf(xjv[f], xjv[f], nj);

    // ---- WMMA Gram tile: dot[M][N] = sum_k X[rowbase+M][k] * X[colbase+N][k]
    // A 16x4 f32 fragment: lane holds M=lane&15, K-pair at kb + 2*half.
    // B 4x16 f32 fragment: lane holds N=lane&15, same K-pair -> identical load.
    const int mA  = rowbase + n;
    const int mAc = (mA < V_) ? mA : (V_ - 1);
    v8f dot = {};
#pragma unroll
    for (int kb = 0; kb < F_; kb += 4) {
        const int k = kb + half * 2;
        v2f af, bf;
        af.x = xs[(size_t)mAc * F_ + k];
        af.y = xs[(size_t)mAc * F_ + k + 1];
        bf.x = xs[(size_t)jc  * F_ + k];
        bf.y = xs[(size_t)jc  * F_ + k + 1];
        dot = __builtin_amdgcn_wmma_f32_16x16x4_f32(
            /*neg_a=*/false, af, /*neg_b=*/false, bf,
            /*c_mod=*/(short)0, dot, /*reuse_a=*/false, /*reuse_b=*/false);
    }

    // ---- per-row work: w, exp, stores, half-wave reductions
#pragma unroll
    for (int r = 0; r < 8; ++r) {
        const int mi  = rowbase + r + half * 8;   // C layout: VGPR r, lanes>=16 -> M+8
        const int mic = (mi < V_) ? mi : (V_ - 1);
        const float* __restrict__ xi = xs + (size_t)mic * F_;

        float w = 0.f, ni = 0.f;
#pragma unroll
        for (int f = 0; f < F_; ++f) {
            const float xv = xi[f];
            w  = fmaf(fabsf(xv - xjv[f]), av[f], w);
            ni = fmaf(xv, xv, ni);
        }

        const bool valid = (mi < V_) && (j < V_);
        const float t  = valid ? expf(w) : 0.f;          // masked exp score
        const float ds = ni + nj - 2.f * dot[r];         // ||x_i - x_j||^2

        if (valid)
            Snum[(size_t)slab * V_ * V_ + (size_t)mi * V_ + j] = t;

        // reduce over the 16 lanes sharing this row (width-16 xor shuffles)
        float s0 = t, s1 = t * ds, s2 = t * t;
#pragma unroll
        for (int m = 8; m >= 1; m >>= 1) {
            s0 += __shfl_xor(s0, m, 16);
            s1 += __shfl_xor(s1, m, 16);
            s2 += __shfl_xor(s2, m, 16);
        }
        if (n == 0 && mi < V_) {
            atomicAdd(&rowsum[slab * V_ + mi], s0);
            atomicAdd(&numer [slab * V_ + mi], s1);
            atomicAdd(&sqsum [slab * V_ + mi], s2);
        }
    }
}

// ---------------- pass 2: L1-normalize S in place --------------------------
__global__ void gl_norm_kernel(float* __restrict__ S,
                               const float* __restrict__ rowsum) {
    const size_t total = (size_t)SLABS * V_ * V_;
    size_t idx = (size_t)blockIdx.x * blockDim.x + threadIdx.x;
    if (idx >= total) return;
    const size_t row = idx / V_;                 // = slab*V_ + i
    const float rs = fmaxf(rowsum[row], EPS_);
    S[idx] = S[idx] / rs;
}

// ---------------- pass 3: fold per-row partials into the two scalars -------
__global__ void gl_reduce_kernel(const float* __restrict__ rowsum,
                                 const float* __restrict__ numer,
                                 const float* __restrict__ sqsum,
                                 float* __restrict__ acc) {
    int i = blockIdx.x * blockDim.x + threadIdx.x;
    if (i >= NROWS) return;
    const float rs = fmaxf(rowsum[i], EPS_);
    // diff_loss = sum_t mean_b (.) = (1/B) * sum over all rows of numer/rs
    atomicAdd(&acc[0], (numer[i] / rs) * (1.0f / (float)B_));
    // f_norm_loss = alpha * sum S^2 = alpha * sum_rows sqsum / rs^2
    atomicAdd(&acc[1], ALPHA * sqsum[i] / (rs * rs));
}

extern "C" void kernel_launch(void* const* d_in, const int* in_sizes, int n_in,
                              void* d_out, int out_size, void* d_ws, size_t ws_size,
                              hipStream_t stream) {
    const float* x = (const float*)d_in[0];   // [4,12,207,32]
    const float* a = (const float*)d_in[1];   // [32]

    const size_t NS = (size_t)SLABS * V_ * V_;   // 2,056,752 S elements
    float* S   = (float*)d_out;                  // S first, then 2 scalars
    float* acc = (float*)d_out + NS;             // [diff_loss, f_norm_loss]

    float* ws     = (float*)d_ws;
    float* rowsum = ws;
    float* numer  = ws + NROWS;
    float* sqsum  = ws + 2 * NROWS;

    gl_init_kernel<<<(NROWS + 255) / 256, 256, 0, stream>>>(rowsum, numer, sqsum, acc);

    gl_tile_kernel<<<SLABS * VT * VT, 32, 0, stream>>>(x, a, S, rowsum, numer, sqsum);

    gl_norm_kernel<<<(int)((NS + 255) / 256), 256, 0, stream>>>(S, rowsum);

    gl_reduce_kernel<<<(NROWS + 255) / 256, 256, 0, stream>>>(rowsum, numer, sqsum, acc);
}